// BiCG_Net_73358041416144
// MI455X (gfx1250) — compile-verified
//
#include <hip/hip_runtime.h>
#include <math.h>

// BiCGSTAB 5-point stencil solver, B=8 independent 384x384 fp32 systems.
// Fixed 30-step kernel graph; per-batch branch flags live on device.
// Reductions use V_WMMA_F32_16X16X4_F32 as an exact fp32 horizontal-sum.

#define NXI   384
#define NPTS  (NXI * NXI)        // 147456
#define NB    8
#define TPB   256
#define NBLK  (NPTS / TPB)       // 576 blocks per batch
#define KMAX_C 30
#define EPS_C 1e-9f

typedef float v2f __attribute__((ext_vector_type(2)));
typedef float v8f __attribute__((ext_vector_type(8)));

struct BState {
  float b_val, r0_abs, r_abs, sigma, v_abs, alpha, omega, beta, rr0_old, ssq;
  int   k, CONV, R1, NR, C3, C4;
};

__device__ __forceinline__ float fro_of(float sq) { return sq > 0.f ? sqrtf(sq) : 0.f; }
__device__ __forceinline__ float thr_c() { return EPS_C * (float)NPTS; }  // EPS*NX*NY

// ---- exact 32-lane fp32 sum via V_WMMA_F32_16X16X4_F32 -----------------
// A holds one value per lane (a.y = 0), B = all-ones, C = 0.
// D[m,n] = rowsum_m of A.  Sum of the 8 D regs on lane L covers half the
// rows; adding the shfl_xor(16) partner covers all 16 rows => total sum of
// every A element = sum over lanes. Layout-robust: B is uniform 1.0.
__device__ __forceinline__ float wave_sum_wmma(float v) {
  v2f a; a.x = v;  a.y = 0.f;
  v2f b; b.x = 1.f; b.y = 1.f;
  v8f c = {};
  v8f d = __builtin_amdgcn_wmma_f32_16x16x4_f32(false, a, false, b, (short)0, c, false, false);
  float h = ((d[0] + d[1]) + (d[2] + d[3])) + ((d[4] + d[5]) + (d[6] + d[7]));
  h += __shfl_xor(h, 16, 32);
  return h;  // identical across all 32 lanes
}

// Deterministic 256-thread block sum. Result valid in thread 0.
__device__ __forceinline__ float block_sum256(float v, float* sh8) {
  float t = wave_sum_wmma(v);
  int lane = threadIdx.x & 31;
  int wid  = threadIdx.x >> 5;
  if (lane == 0) sh8[wid] = t;
  __syncthreads();
  float s = 0.f;
  if (threadIdx.x == 0) {
#pragma unroll
    for (int i = 0; i < 8; ++i) s += sh8[i];
  }
  __syncthreads();  // allow shared reuse by the next call
  return s;
}

// ---- 5-point stencil with symmetric (clamped) boundaries ----------------
__device__ __forceinline__ float applyA(const float* __restrict__ boo, const float* __restrict__ bmo,
                                        const float* __restrict__ bom, const float* __restrict__ bop,
                                        const float* __restrict__ bpo, const float* __restrict__ a,
                                        size_t base, int idx) {
  int i = idx / NXI, j = idx - i * NXI;
  int im = (i > 0)       ? idx - NXI : idx;
  int ip = (i < NXI - 1) ? idx + NXI : idx;
  int jm = (j > 0)       ? idx - 1   : idx;
  int jp = (j < NXI - 1) ? idx + 1   : idx;
  const float* A = a + base;
  return boo[base + idx] * A[idx] + bmo[base + idx] * A[im] + bom[base + idx] * A[jm]
       + bop[base + idx] * A[jp] + bpo[base + idx] * A[ip];
}

// symmetric-pad index map: padded t in [0,386) -> source index in [0,384)
__device__ __forceinline__ int refl(int t) { return t == 0 ? 0 : (t == 385 ? 383 : t - 1); }

// ======================= init kernels ===================================
__global__ __launch_bounds__(TPB) void k_mean(const float* __restrict__ V, float* __restrict__ part) {
  int b = blockIdx.y;
  int idx = blockIdx.x * TPB + threadIdx.x;
  __shared__ float sh[8];
  float s = block_sum256(V[(size_t)b * NPTS + idx], sh);
  if (threadIdx.x == 0) part[(0 * NB + b) * NBLK + blockIdx.x] = s;
}

__global__ void k_init_scalars(const float* __restrict__ part, BState* __restrict__ st) {
  int b = threadIdx.x;
  if (b >= NB) return;
  float s = 0.f;
  for (int i = 0; i < NBLK; ++i) s += part[(0 * NB + b) * NBLK + i];
  BState q = {};
  q.b_val = s / (float)NPTS + 1.0f;  // OFFSET; U and b are this constant
  st[b] = q;
}

__global__ __launch_bounds__(TPB) void k_stencil_init(
    const float* __restrict__ V, const float* __restrict__ M1, const float* __restrict__ M2,
    float* __restrict__ boo, float* __restrict__ bmo, float* __restrict__ bom,
    float* __restrict__ bop, float* __restrict__ bpo, float* __restrict__ x,
    const BState* __restrict__ st) {
  int b = blockIdx.y;
  size_t base = (size_t)b * NPTS;
  int idx = blockIdx.x * TPB + threadIdx.x;
  int i = idx / NXI, j = idx - i * NXI;
  int I = i + 1, J = j + 1;  // padded/transposed coords
  const float* Vb = V + base;
  const float* m1b = M1 + base;
  const float* m2b = M2 + base;
  // Vp[I,J] = sym_pad(V) transposed + OFFSET  =>  V[refl(J), refl(I)] + 1
#define VP(II, JJ)  (Vb[refl(JJ) * NXI + refl(II)] + 1.0f)
#define MM1(II, JJ) (m1b[refl(JJ) * NXI + refl(II)])
#define MM2(II, JJ) (m2b[refl(JJ) * NXI + refl(II)])
  float vc  = VP(I, J);
  float vxp = VP(I + 1, J), vxm = VP(I - 1, J);
  float vyp = VP(I, J + 1), vym = VP(I, J - 1);
  float d1c = ((vxp - vc) / (0.5f * (vxp + vc))) * MM1(I, J);       // d1[I,J]
  float d1m = ((vc - vxm) / (0.5f * (vc + vxm))) * MM1(I - 1, J);   // d1[I-1,J]
  float d2c = ((vyp - vc) / (0.5f * (vyp + vc))) * MM2(I, J);       // d2[I,J]
  float d2m = ((vc - vym) / (0.5f * (vc + vym))) * MM2(I, J - 1);   // d2[I,J-1]
#undef VP
#undef MM1
#undef MM2
  const float rx = 5.0f, ry = 5.0f, rxx = 10.0f, ryy = 10.0f;  // TAU=10, HX=HY=1
  boo[base + idx] = 1.0f + 2.0f * (rxx + ryy) - rx * (d1m - d1c) - ry * (d2m - d2c);
  bpo[base + idx] = -rxx + rx * d1c;
  bmo[base + idx] = -rxx - rx * d1m;
  bop[base + idx] = -ryy + ry * d2c;
  bom[base + idx] = -ryy - ry * d2m;
  x[base + idx] = st[b].b_val;  // x0 = U (constant per batch)
}

__global__ __launch_bounds__(TPB) void k_resid_init(
    const float* __restrict__ boo, const float* __restrict__ bmo, const float* __restrict__ bom,
    const float* __restrict__ bop, const float* __restrict__ bpo, const float* __restrict__ x,
    float* __restrict__ p, float* __restrict__ r, float* __restrict__ r0v,
    float* __restrict__ part, const BState* __restrict__ st) {
  int b = blockIdx.y;
  size_t base = (size_t)b * NPTS;
  int idx = blockIdx.x * TPB + threadIdx.x;
  float pv = st[b].b_val - applyA(boo, bmo, bom, bop, bpo, x, base, idx);
  p[base + idx] = pv; r[base + idx] = pv; r0v[base + idx] = pv;
  __shared__ float sh[8];
  float s = block_sum256(pv * pv, sh);
  if (threadIdx.x == 0) part[(0 * NB + b) * NBLK + blockIdx.x] = s;
}

__global__ void k_init_finalize(const float* __restrict__ part, BState* __restrict__ st) {
  int b = threadIdx.x;
  if (b >= NB) return;
  float s = 0.f;
  for (int i = 0; i < NBLK; ++i) s += part[(0 * NB + b) * NBLK + i];
  float fa = fro_of(s);
  st[b].r0_abs = fa; st[b].r_abs = fa; st[b].k = 0;
  st[b].CONV = (fa > thr_c()) ? 1 : 0;
}

// ======================= per-step kernels ================================
__global__ __launch_bounds__(TPB) void k_v(
    const float* __restrict__ boo, const float* __restrict__ bmo, const float* __restrict__ bom,
    const float* __restrict__ bop, const float* __restrict__ bpo, const float* __restrict__ p,
    const float* __restrict__ r0v, const float* __restrict__ r, float* __restrict__ v,
    float* __restrict__ part, const BState* __restrict__ st) {
  int b = blockIdx.y;
  size_t base = (size_t)b * NPTS;
  int idx = blockIdx.x * TPB + threadIdx.x;
  float q0 = 0.f, q1 = 0.f, q2 = 0.f;
  if (st[b].CONV) {  // block-uniform branch; EXEC is all-ones at the WMMA below
    float vv = applyA(boo, bmo, bom, bop, bpo, p, base, idx);
    v[base + idx] = vv;
    float rz = r0v[base + idx];
    q0 = vv * rz; q1 = vv * vv; q2 = r[base + idx] * rz;
  }
  __shared__ float sh[8];
  float s0 = block_sum256(q0, sh);
  float s1 = block_sum256(q1, sh);
  float s2 = block_sum256(q2, sh);
  if (threadIdx.x == 0) {
    part[(0 * NB + b) * NBLK + blockIdx.x] = s0;
    part[(1 * NB + b) * NBLK + blockIdx.x] = s1;
    part[(2 * NB + b) * NBLK + blockIdx.x] = s2;
  }
}

__global__ void k_v_reduce(const float* __restrict__ part, BState* __restrict__ st) {
  int b = threadIdx.x;
  if (b >= NB) return;
  BState s = st[b];
  if (s.CONV) {
    float sg = 0.f, vq = 0.f, rr = 0.f;
    for (int i = 0; i < NBLK; ++i) {
      sg += part[(0 * NB + b) * NBLK + i];
      vq += part[(1 * NB + b) * NBLK + i];
      rr += part[(2 * NB + b) * NBLK + i];
    }
    s.sigma = sg; s.v_abs = fro_of(vq); s.rr0_old = rr;
    int RES = (sg <= EPS_C * s.v_abs * s.r0_abs) ? 1 : 0;
    s.R1 = RES; s.NR = 1 - RES;
    if (s.R1) s.k += 1;
    if (s.NR) s.alpha = rr / sg;  // alpha = <r,r0>/sigma
  } else { s.R1 = 0; s.NR = 0; }
  st[b] = s;
}

__global__ __launch_bounds__(TPB) void k_restart(
    const float* __restrict__ boo, const float* __restrict__ bmo, const float* __restrict__ bom,
    const float* __restrict__ bop, const float* __restrict__ bpo, const float* __restrict__ x,
    float* __restrict__ p, float* __restrict__ r, float* __restrict__ r0v,
    float* __restrict__ part, const BState* __restrict__ st) {
  int b = blockIdx.y;
  size_t base = (size_t)b * NPTS;
  int idx = blockIdx.x * TPB + threadIdx.x;
  float q0 = 0.f;
  if (st[b].R1) {
    float pv = st[b].b_val - applyA(boo, bmo, bom, bop, bpo, x, base, idx);
    p[base + idx] = pv; r[base + idx] = pv; r0v[base + idx] = pv;
    q0 = pv * pv;
  }
  __shared__ float sh[8];
  float s = block_sum256(q0, sh);
  if (threadIdx.x == 0) part[(0 * NB + b) * NBLK + blockIdx.x] = s;
}

__global__ void k_restart_reduce(const float* __restrict__ part, BState* __restrict__ st) {
  int b = threadIdx.x;
  if (b >= NB) return;
  if (st[b].R1) {
    float s = 0.f;
    for (int i = 0; i < NBLK; ++i) s += part[(0 * NB + b) * NBLK + i];
    float fa = fro_of(s);
    st[b].r0_abs = fa; st[b].r_abs = fa;
  }
}

__global__ __launch_bounds__(TPB) void k_s(
    const float* __restrict__ r, const float* __restrict__ v, float* __restrict__ sarr,
    float* __restrict__ part, const BState* __restrict__ st) {
  int b = blockIdx.y;
  size_t base = (size_t)b * NPTS;
  int idx = blockIdx.x * TPB + threadIdx.x;
  float q0 = 0.f;
  if (st[b].NR) {
    float sv = r[base + idx] - st[b].alpha * v[base + idx];
    sarr[base + idx] = sv;
    q0 = sv * sv;
  }
  __shared__ float sh[8];
  float s = block_sum256(q0, sh);
  if (threadIdx.x == 0) part[(0 * NB + b) * NBLK + blockIdx.x] = s;
}

__global__ void k_s_reduce(const float* __restrict__ part, BState* __restrict__ st) {
  int b = threadIdx.x;
  if (b >= NB) return;
  BState s = st[b];
  if (s.NR) {
    float sq = 0.f;
    for (int i = 0; i < NBLK; ++i) sq += part[(0 * NB + b) * NBLK + i];
    s.ssq = sq;
    int C2 = (fro_of(sq) <= thr_c()) ? 1 : 0;
    s.C3 = C2; s.C4 = 1 - C2;
  } else { s.C3 = 0; s.C4 = 0; }
  st[b] = s;
}

__global__ __launch_bounds__(TPB) void k_t(
    const float* __restrict__ boo, const float* __restrict__ bmo, const float* __restrict__ bom,
    const float* __restrict__ bop, const float* __restrict__ bpo, const float* __restrict__ sarr,
    float* __restrict__ tarr, float* __restrict__ x, float* __restrict__ r,
    const float* __restrict__ p, float* __restrict__ part, const BState* __restrict__ st) {
  int b = blockIdx.y;
  size_t base = (size_t)b * NPTS;
  int idx = blockIdx.x * TPB + threadIdx.x;
  float q0 = 0.f, q1 = 0.f;
  if (st[b].C4) {
    float tv = applyA(boo, bmo, bom, bop, bpo, sarr, base, idx);
    tarr[base + idx] = tv;
    q0 = tv * tv; q1 = tv * sarr[base + idx];
  } else if (st[b].C3) {  // converged: x += alpha*p ; r = s
    x[base + idx] += st[b].alpha * p[base + idx];
    r[base + idx] = sarr[base + idx];
  }
  __shared__ float sh[8];
  float s0 = block_sum256(q0, sh);
  float s1 = block_sum256(q1, sh);
  if (threadIdx.x == 0) {
    part[(0 * NB + b) * NBLK + blockIdx.x] = s0;
    part[(1 * NB + b) * NBLK + blockIdx.x] = s1;
  }
}

__global__ void k_t_reduce(const float* __restrict__ part, BState* __restrict__ st) {
  int b = threadIdx.x;
  if (b >= NB) return;
  BState s = st[b];
  if (s.C4) {
    float tt = 0.f, ts = 0.f;
    for (int i = 0; i < NBLK; ++i) {
      tt += part[(0 * NB + b) * NBLK + i];
      ts += part[(1 * NB + b) * NBLK + i];
    }
    s.omega = ts / tt;
    st[b] = s;
  }
}

__global__ __launch_bounds__(TPB) void k_x_r(
    const float* __restrict__ sarr, const float* __restrict__ tarr, const float* __restrict__ p,
    const float* __restrict__ r0v, float* __restrict__ x, float* __restrict__ r,
    float* __restrict__ part, const BState* __restrict__ st) {
  int b = blockIdx.y;
  size_t base = (size_t)b * NPTS;
  int idx = blockIdx.x * TPB + threadIdx.x;
  float q0 = 0.f, q1 = 0.f;
  if (st[b].C4) {
    float al = st[b].alpha, om = st[b].omega;
    x[base + idx] += al * p[base + idx] + om * sarr[base + idx];
    float rn = sarr[base + idx] - om * tarr[base + idx];
    r[base + idx] = rn;
    q0 = rn * r0v[base + idx]; q1 = rn * rn;
  }
  __shared__ float sh[8];
  float s0 = block_sum256(q0, sh);
  float s1 = block_sum256(q1, sh);
  if (threadIdx.x == 0) {
    part[(0 * NB + b) * NBLK + blockIdx.x] = s0;
    part[(1 * NB + b) * NBLK + blockIdx.x] = s1;
  }
}

__global__ void k_final_reduce(const float* __restrict__ part, BState* __restrict__ st) {
  int b = threadIdx.x;
  if (b >= NB) return;
  BState s = st[b];
  if (s.C4) {
    float rr = 0.f, rq = 0.f;
    for (int i = 0; i < NBLK; ++i) {
      rr += part[(0 * NB + b) * NBLK + i];
      rq += part[(1 * NB + b) * NBLK + i];
    }
    // den = <r_old, r0> == rr0_old computed for alpha this step
    s.beta  = (s.alpha / s.omega) * (rr / s.rr0_old);
    s.r_abs = fro_of(rq);
  }
  if (s.C3) s.r_abs = fro_of(s.ssq);  // r = s
  if (s.NR) s.k += 1;
  s.CONV = ((s.k < KMAX_C) && (s.r_abs > thr_c())) ? 1 : 0;  // gate for next step
  st[b] = s;
}

__global__ __launch_bounds__(TPB) void k_p(
    const float* __restrict__ r, const float* __restrict__ v, float* __restrict__ p,
    const BState* __restrict__ st) {
  int b = blockIdx.y;
  size_t base = (size_t)b * NPTS;
  int idx = blockIdx.x * TPB + threadIdx.x;
  if (st[b].C4) {
    p[base + idx] = r[base + idx] + st[b].beta * (p[base + idx] - st[b].omega * v[base + idx]);
  }
}

// ======================= launcher =======================================
extern "C" void kernel_launch(void* const* d_in, const int* in_sizes, int n_in,
                              void* d_out, int out_size, void* d_ws, size_t ws_size,
                              hipStream_t stream) {
  (void)in_sizes; (void)n_in; (void)out_size; (void)ws_size;
  const float* V  = (const float*)d_in[0];
  const float* M1 = (const float*)d_in[1];
  const float* M2 = (const float*)d_in[2];
  float* x = (float*)d_out;  // output IS the solution interior; used as state

  char* ws = (char*)d_ws;
  BState* st  = (BState*)ws;                      // 8 * 64 B
  float* part = (float*)(ws + 1024);              // 3 * 8 * 576 floats
  float* arr  = (float*)(ws + 1024 + 65536);      // 11 arrays of B*NPTS floats
  const size_t AN = (size_t)NB * NPTS;
  float* boo  = arr + 0 * AN;
  float* bmo  = arr + 1 * AN;
  float* bom  = arr + 2 * AN;
  float* bop  = arr + 3 * AN;
  float* bpo  = arr + 4 * AN;
  float* r0v  = arr + 5 * AN;
  float* r    = arr + 6 * AN;
  float* p    = arr + 7 * AN;
  float* v    = arr + 8 * AN;
  float* sarr = arr + 9 * AN;
  float* tarr = arr + 10 * AN;

  dim3 grid(NBLK, NB), blk(TPB);

  k_mean<<<grid, blk, 0, stream>>>(V, part);
  k_init_scalars<<<1, 32, 0, stream>>>(part, st);
  k_stencil_init<<<grid, blk, 0, stream>>>(V, M1, M2, boo, bmo, bom, bop, bpo, x, st);
  k_resid_init<<<grid, blk, 0, stream>>>(boo, bmo, bom, bop, bpo, x, p, r, r0v, part, st);
  k_init_finalize<<<1, 32, 0, stream>>>(part, st);

  for (int it = 0; it < KMAX_C; ++it) {
    k_v<<<grid, blk, 0, stream>>>(boo, bmo, bom, bop, bpo, p, r0v, r, v, part, st);
    k_v_reduce<<<1, 32, 0, stream>>>(part, st);
    k_restart<<<grid, blk, 0, stream>>>(boo, bmo, bom, bop, bpo, x, p, r, r0v, part, st);
    k_restart_reduce<<<1, 32, 0, stream>>>(part, st);
    k_s<<<grid, blk, 0, stream>>>(r, v, sarr, part, st);
    k_s_reduce<<<1, 32, 0, stream>>>(part, st);
    k_t<<<grid, blk, 0, stream>>>(boo, bmo, bom, bop, bpo, sarr, tarr, x, r, p, part, st);
    k_t_reduce<<<1, 32, 0, stream>>>(part, st);
    k_x_r<<<grid, blk, 0, stream>>>(sarr, tarr, p, r0v, x, r, part, st);
    k_final_reduce<<<1, 32, 0, stream>>>(part, st);
    k_p<<<grid, blk, 0, stream>>>(r, v, p, st);
  }
}